// TrafficGCN_77721728188708
// MI455X (gfx1250) — compile-verified
//
#include <hip/hip_runtime.h>
#include <hip/hip_bf16.h>

typedef __attribute__((ext_vector_type(2))) float v2f;
typedef __attribute__((ext_vector_type(8))) float v8f;

__device__ __forceinline__ void atomicAddF32(float* p, float v) {
    // agent-scope relaxed fp add -> global_atomic_add_f32 (no CAS loop)
    __hip_atomic_fetch_add(p, v, __ATOMIC_RELAXED, __HIP_MEMORY_SCOPE_AGENT);
}

// ---- K1: deg[i] = 1 (self loop) ----
__global__ void k_init_deg(float* __restrict__ deg, int n) {
    int i = blockIdx.x * blockDim.x + threadIdx.x;
    if (i < n) deg[i] = 1.0f;
}

// ---- K2: deg[dst[e]] += 1 ----
__global__ void k_accum_deg(const int* __restrict__ dst, float* __restrict__ deg, int E) {
    int e = blockIdx.x * blockDim.x + threadIdx.x;
    if (e < E) atomicAddF32(deg + dst[e], 1.0f);
}

// ---- K3: inv[i] = rsqrt(deg[i])  (deg >= 1 always) ----
__global__ void k_inv_sqrt(const float* __restrict__ deg, float* __restrict__ inv, int n) {
    int i = blockIdx.x * blockDim.x + threadIdx.x;
    if (i < n) inv[i] = rsqrtf(deg[i]);
}

// ---- K4: hlin[N,16] = x[N,3] @ W1[3,16] via V_WMMA_F32_16X16X4_F32 ----
// One wave handles a 16-node tile. A = 16x4 (3 feats + zero pad), B = 4x16.
// A layout: lanes 0-15 -> M=lane, VGPR0=K0 VGPR1=K1; lanes 16-31 -> M=lane-16, K2/K3.
// B layout (mirror): lanes 0-15 -> N=lane, K0/K1; lanes 16-31 -> K2/K3.
// D layout: VGPR v -> row (v + 8*half), col = lane&15.
// Branchless A/B setup (v_cndmask, EXEC untouched) + wave-uniform full-tile
// store fast path (8 unconditional b32 stores, no EXEC toggling).
__global__ void k_linear1_wmma(const float* __restrict__ x, const float* __restrict__ W1,
                               float* __restrict__ hlin, int n) {
    int wave = (blockIdx.x * blockDim.x + threadIdx.x) >> 5;
    int lane = threadIdx.x & 31;
    int base = wave * 16;
    if (base >= n) return;               // wave-uniform: EXEC stays all-ones
    bool hi  = lane >= 16;               // hi half covers K=2,3
    int  idx = lane & 15;

    int m  = base + idx;
    int mc = m < n ? m : (n - 1);        // clamp loads for partial tile
    float x0 = x[mc * 3 + 0], x1 = x[mc * 3 + 1], x2 = x[mc * 3 + 2];
    float w0 = W1[idx], w1 = W1[16 + idx], w2 = W1[32 + idx];

    v2f a, b;
    a.x = hi ? x2 : x0;  a.y = hi ? 0.0f : x1;   // K=3 column zero-padded
    b.x = hi ? w2 : w0;  b.y = hi ? 0.0f : w1;

    v8f c = {};
    c = __builtin_amdgcn_wmma_f32_16x16x4_f32(false, a, false, b, (short)0, c, false, false);

    int rowoff = hi ? 8 : 0;
    if (base + 16 <= n) {                // wave-uniform full-tile fast path
        float* o = hlin + (size_t)(base + rowoff) * 16 + idx;
#pragma unroll
        for (int v = 0; v < 8; ++v) o[v * 16] = c[v];
    } else {                             // last partial tile
#pragma unroll
        for (int v = 0; v < 8; ++v) {
            int mm = base + v + rowoff;
            if (mm < n) hlin[mm * 16 + idx] = c[v];
        }
    }
}

// ---- K5: acc1[i][k] = b1[k] + inv[i]^2 * hlin[i][k]  (self-loop term) ----
__global__ void k_init_acc1(const float* __restrict__ hlin, const float* __restrict__ b1,
                            const float* __restrict__ inv, float* __restrict__ acc1, int n16) {
    int t = blockIdx.x * blockDim.x + threadIdx.x;
    if (t >= n16) return;
    int i = t >> 4, k = t & 15;
    float iv = inv[i];
    acc1[t] = b1[k] + iv * iv * hlin[t];
}

// ---- K6: layer-1 edge scatter: acc1[dst] += inv[s]*inv[d] * hlin[src]  (16 thr/edge) ----
__global__ void k_scatter1(const int* __restrict__ src, const int* __restrict__ dst,
                           const float* __restrict__ inv, const float* __restrict__ hlin,
                           float* __restrict__ acc1, unsigned total /* E*16 */) {
    unsigned t = blockIdx.x * blockDim.x + threadIdx.x;
    if (t >= total) return;
    int e = (int)(t >> 4), k = (int)(t & 15u);
    int s = src[e], d = dst[e];
    float coeff = inv[s] * inv[d];
    atomicAddF32(acc1 + d * 16 + k, coeff * hlin[s * 16 + k]);
}

// ---- K7: g[i] = relu(acc1[i]) @ W2 ; out[i] = b2 + inv[i]^2 * g[i] ----
__global__ void k_linear2(const float* __restrict__ acc1, const float* __restrict__ W2,
                          const float* __restrict__ b2, const float* __restrict__ inv,
                          float* __restrict__ g, float* __restrict__ out, int n) {
    int i = blockIdx.x * blockDim.x + threadIdx.x;
    if (i >= n) return;
    const float4* h = (const float4*)(acc1 + i * 16);
    const float4* w = (const float4*)W2;
    float s = 0.0f;
#pragma unroll
    for (int c4 = 0; c4 < 4; ++c4) {
        float4 hv = h[c4];
        float4 wv = w[c4];
        s += fmaxf(hv.x, 0.0f) * wv.x + fmaxf(hv.y, 0.0f) * wv.y +
             fmaxf(hv.z, 0.0f) * wv.z + fmaxf(hv.w, 0.0f) * wv.w;
    }
    g[i] = s;
    float iv = inv[i];
    out[i] = b2[0] + iv * iv * s;
}

// ---- K8: layer-2 edge scatter: out[dst] += inv[s]*inv[d] * g[src] ----
__global__ void k_scatter2(const int* __restrict__ src, const int* __restrict__ dst,
                           const float* __restrict__ inv, const float* __restrict__ g,
                           float* __restrict__ out, int E) {
    int e = blockIdx.x * blockDim.x + threadIdx.x;
    if (e >= E) return;
    int s = src[e], d = dst[e];
    atomicAddF32(out + d, inv[s] * inv[d] * g[s]);
}

extern "C" void kernel_launch(void* const* d_in, const int* in_sizes, int n_in,
                              void* d_out, int out_size, void* d_ws, size_t ws_size,
                              hipStream_t stream) {
    const float* x    = (const float*)d_in[0];   // [N,3]
    const int*   ei   = (const int*)d_in[1];     // [2,E]
    const float* W1   = (const float*)d_in[2];   // [3,16]
    const float* b1   = (const float*)d_in[3];   // [16]
    const float* W2   = (const float*)d_in[4];   // [16,1]
    const float* b2   = (const float*)d_in[5];   // [1]
    float* out = (float*)d_out;                  // [N]

    const int N = in_sizes[0] / 3;
    const int E = in_sizes[1] / 2;
    const int* src = ei;
    const int* dst = ei + E;

    // workspace layout (floats): deg[N] | inv[N] | hlin[N*16] | acc1[N*16] | g[N]
    float* deg  = (float*)d_ws;
    float* inv  = deg  + N;
    float* hlin = inv  + N;
    float* acc1 = hlin + (size_t)N * 16;
    float* g    = acc1 + (size_t)N * 16;

    const int B = 256;
    // degrees
    k_init_deg <<<(N + B - 1) / B, B, 0, stream>>>(deg, N);
    k_accum_deg<<<(E + B - 1) / B, B, 0, stream>>>(dst, deg, E);
    k_inv_sqrt <<<(N + B - 1) / B, B, 0, stream>>>(deg, inv, N);
    // layer 1 linear (WMMA): 8 waves/block * 16 nodes/wave = 128 nodes/block
    k_linear1_wmma<<<(N + 127) / 128, B, 0, stream>>>(x, W1, hlin, N);
    // self-loop init + edge scatter
    int n16 = N * 16;
    k_init_acc1<<<(n16 + B - 1) / B, B, 0, stream>>>(hlin, b1, inv, acc1, n16);
    unsigned total1 = (unsigned)E * 16u;
    k_scatter1<<<(total1 + B - 1) / B, B, 0, stream>>>(src, dst, inv, hlin, acc1, total1);
    // layer 2: relu + linear + self-loop, then edge scatter
    k_linear2<<<(N + B - 1) / B, B, 0, stream>>>(acc1, W2, b2, inv, g, out, N);
    k_scatter2<<<(E + B - 1) / B, B, 0, stream>>>(src, dst, inv, g, out, E);
}